// TorchIndex_48352741818800
// MI455X (gfx1250) — compile-verified
//
#include <hip/hip_runtime.h>
#include <hip/hip_bf16.h>

// Problem constants (match reference)
#define BQ     64        // queries
#define DD     128       // dim
#define NN     1000000   // vectors
#define KK     100       // top-k
#define TILE_N 32        // vectors per inner tile (2 wmma sub-tiles of 16)
#define CHUNK  4096      // vectors per block
#define NB_BLK ((NN + CHUNK - 1) / CHUNK)   // 245
#define NBINS  256       // histogram bins, covers [-128,128)
#define QSTR   132       // padded LDS row stride (floats): 16B-aligned rows, conflict-free
#define CAP    4096      // max candidates kept per query
#define SORTN  4096      // bitonic sort size (== CAP)

#if __has_builtin(__builtin_amdgcn_global_load_async_to_lds_b128)
#define HAVE_ASYNC_LDS 1
#else
#define HAVE_ASYNC_LDS 0
#endif

typedef float v2f __attribute__((ext_vector_type(2)));
typedef float v8f __attribute__((ext_vector_type(8)));
typedef int   v4i __attribute__((ext_vector_type(4)));
typedef __attribute__((address_space(1))) v4i as1_v4i;
typedef __attribute__((address_space(3))) v4i as3_v4i;

__device__ __forceinline__ int score_bin(float s) {
    int b = (int)floorf(s) + 128;
    return b < 0 ? 0 : (b > (NBINS - 1) ? (NBINS - 1) : b);
}

__device__ __forceinline__ unsigned ordered_key(float s) {
    unsigned b = __float_as_uint(s);
    return (b & 0x80000000u) ? ~b : (b | 0x80000000u);
}

__device__ __forceinline__ float key_to_float(unsigned k) {
    unsigned b = (k & 0x80000000u) ? (k ^ 0x80000000u) : ~k;
    return __uint_as_float(b);
}

__global__ void zero_kernel(unsigned* p, int n) {
    int i = blockIdx.x * blockDim.x + threadIdx.x;
    if (i < n) p[i] = 0u;
}

// PHASE 0: compute scores via WMMA, build per-query histogram.
// PHASE 1: recompute scores, collect candidates >= threshold bin.
template <int PHASE>
__global__ __launch_bounds__(256) void score_kernel(
    const float* __restrict__ query, const float* __restrict__ vectors,
    unsigned* __restrict__ g_hist, const int* __restrict__ g_thr,
    unsigned* __restrict__ g_cnt, unsigned long long* __restrict__ g_cand)
{
    __shared__ float    qlds[BQ * QSTR];            // 64x128 queries, padded
    __shared__ float    vlds[2][TILE_N * QSTR];     // double-buffered 32x128 vector tile
    __shared__ unsigned sel[BQ * NBINS];            // PHASE0: hist; PHASE1: [0..63] thresholds

    const int tid  = threadIdx.x;
    const int lane = tid & 31;
    const int wave = tid >> 5;
    const int m16  = lane & 15;     // row within 16 (M for A, N for B)
    const int h    = lane >> 4;     // half-wave select
    const int qt   = wave & 3;      // query tile 0..3  (16 queries each)
    const int ns   = wave >> 2;     // n sub-tile 0..1  (16 vectors each)
    const long long n0 = (long long)blockIdx.x * CHUNK;
    const int NT = CHUNK / TILE_N;  // 128 tile iterations

    if (PHASE == 0) {
        for (int i = tid; i < BQ * NBINS; i += 256) sel[i] = 0u;
    } else {
        if (tid < BQ) sel[tid] = (unsigned)g_thr[tid];
    }

    // Stage all 64 queries (coalesced float4 loads)
    for (int i4 = tid; i4 < BQ * (DD / 4); i4 += 256) {
        int r = i4 >> 5, c4 = i4 & 31;
        float4 v = ((const float4*)query)[i4];
        float* dst = &qlds[r * QSTR + c4 * 4];
        dst[0] = v.x; dst[1] = v.y; dst[2] = v.z; dst[3] = v.w;
    }
    __syncthreads();

    // Hoist all A-matrix fragments into registers (queries are loop-invariant).
    v2f afrag[DD / 4];
    {
        const float* qb = &qlds[(qt * 16 + m16) * QSTR];
#pragma unroll
        for (int k0 = 0; k0 < DD; k0 += 4) {
            v2f a; a.x = qb[k0 + 2 * h]; a.y = qb[k0 + 2 * h + 1];
            afrag[k0 >> 2] = a;
        }
    }

    // Stage one 32x128 vector tile into vlds[buf] (async if available).
    auto stage = [&](int tt, int buf) {
        const long long nb = n0 + (long long)tt * TILE_N;
        for (int i4 = tid; i4 < TILE_N * (DD / 4); i4 += 256) {
            int r = i4 >> 5, c4 = i4 & 31;
            long long nv = nb + r;
            if (nv >= NN) nv = NN - 1;   // clamp; out-of-range lanes discarded at consume
#if HAVE_ASYNC_LDS
            __builtin_amdgcn_global_load_async_to_lds_b128(
                (as1_v4i*)(vectors + nv * (long long)DD + (long long)(c4 * 4)),
                (as3_v4i*)&vlds[buf][r * QSTR + c4 * 4], 0, 0);
#else
            float4 v = ((const float4*)vectors)[nv * (long long)(DD / 4) + c4];
            float* dst = &vlds[buf][r * QSTR + c4 * 4];
            dst[0] = v.x; dst[1] = v.y; dst[2] = v.z; dst[3] = v.w;
#endif
        }
    };

    stage(0, 0);

    for (int t = 0; t < NT; ++t) {
        // Issue next tile's copies (redundant re-stage on last iter keeps counts constant)
        int tn = (t + 1 < NT) ? (t + 1) : (NT - 1);
        stage(tn, (t + 1) & 1);
        // Prefetch two tiles ahead (global_prefetch_b8), 8 threads per row
        {
            long long nv2 = n0 + (long long)(t + 2) * TILE_N + (tid >> 3);
            if (nv2 < NN)
                __builtin_prefetch(vectors + nv2 * DD + (size_t)(tid & 7) * 16, 0, 1);
        }
#if HAVE_ASYNC_LDS
        // Drain only the previous tile's 4 per-thread copies; next tile stays in flight.
#if __has_builtin(__builtin_amdgcn_s_wait_asynccnt)
        __builtin_amdgcn_s_wait_asynccnt(4);
#else
        asm volatile("s_wait_asynccnt 4" ::: "memory");
#endif
#endif
        __syncthreads();

        // 16x16 score tile per wave: C += A(16x4) * B(4x16) over k
        const float* vb = &vlds[t & 1][(ns * 16 + m16) * QSTR];
        v8f c = {};
#pragma unroll
        for (int k0 = 0; k0 < DD; k0 += 4) {
            v2f b; b.x = vb[k0 + 2 * h]; b.y = vb[k0 + 2 * h + 1];
            c = __builtin_amdgcn_wmma_f32_16x16x4_f32(
                    false, afrag[k0 >> 2], false, b, (short)0, c, false, false);
        }

        // Consume: lane holds scores for 8 queries at one vector index
        const long long nb = n0 + (long long)t * TILE_N;
        const long long n = nb + ns * 16 + m16;
#pragma unroll
        for (int r = 0; r < 8; ++r) {
            const int q = qt * 16 + 8 * h + r;
            const float s = c[r];
            if (n < NN) {
                const int bin = score_bin(s);
                if (PHASE == 0) {
                    atomicAdd(&sel[q * NBINS + bin], 1u);
                } else {
                    if (bin >= (int)sel[q]) {
                        unsigned pos = atomicAdd(&g_cnt[q], 1u);
                        if (pos < CAP) {
                            unsigned long long e =
                                ((unsigned long long)ordered_key(s) << 32) |
                                (unsigned)(~(unsigned)n);
                            g_cand[(size_t)q * CAP + pos] = e;
                        }
                    }
                }
            }
        }
        __syncthreads();   // compute done before next iteration overwrites other buffer
    }

    if (PHASE == 0) {
        for (int i = tid; i < BQ * NBINS; i += 256) {
            unsigned v = sel[i];
            if (v) atomicAdd(&g_hist[i], v);
        }
    }
}

// Per query: find threshold bin (largest b with suffix count >= K); reset counters.
__global__ void threshold_kernel(const unsigned* __restrict__ g_hist,
                                 int* __restrict__ g_thr, unsigned* __restrict__ g_cnt) {
    int q = threadIdx.x;
    if (q < BQ) {
        unsigned acc = 0; int t = 0;
        for (int b = NBINS - 1; b >= 0; --b) {
            acc += g_hist[q * NBINS + b];
            if (acc >= KK) { t = b; break; }
        }
        g_thr[q] = t;
        g_cnt[q] = 0u;
    }
}

// One block per query: bitonic sort candidates descending, emit top-K.
__global__ __launch_bounds__(256) void topk_kernel(
    const unsigned* __restrict__ g_cnt, const unsigned long long* __restrict__ g_cand,
    float* __restrict__ out_scores, int* __restrict__ out_idx)
{
    __shared__ unsigned long long buf[SORTN];
    const int q = blockIdx.x;
    unsigned cnt = g_cnt[q]; if (cnt > CAP) cnt = CAP;

    for (int i = threadIdx.x; i < SORTN; i += 256)
        buf[i] = (i < (int)cnt) ? g_cand[(size_t)q * CAP + i] : 0ull;
    __syncthreads();

    for (int ksz = 2; ksz <= SORTN; ksz <<= 1) {
        for (int j = ksz >> 1; j > 0; j >>= 1) {
            for (int i = threadIdx.x; i < SORTN; i += 256) {
                int ixj = i ^ j;
                if (ixj > i) {
                    unsigned long long a = buf[i], b = buf[ixj];
                    bool desc = ((i & ksz) == 0);
                    if (desc ? (a < b) : (a > b)) { buf[i] = b; buf[ixj] = a; }
                }
            }
            __syncthreads();
        }
    }

    for (int i = threadIdx.x; i < KK; i += 256) {
        unsigned long long e = buf[i];
        out_scores[q * KK + i] = key_to_float((unsigned)(e >> 32));
        out_idx[q * KK + i]    = (int)(~(unsigned)(e & 0xffffffffu));
    }
}

extern "C" void kernel_launch(void* const* d_in, const int* in_sizes, int n_in,
                              void* d_out, int out_size, void* d_ws, size_t ws_size,
                              hipStream_t stream) {
    const float* query   = (const float*)d_in[0];   // [64,128] f32
    const float* vectors = (const float*)d_in[1];   // [1M,128] f32
    // d_in[2] = k (==100), baked in as KK

    float* out_scores = (float*)d_out;              // [64,100] f32
    int*   out_idx    = (int*)d_out + BQ * KK;      // [64,100] i32

    unsigned char* ws = (unsigned char*)d_ws;
    unsigned*           g_hist = (unsigned*)(ws);                    // 64*256*4 = 65536 B
    int*                g_thr  = (int*)(ws + 65536);                 // 256 B
    unsigned*           g_cnt  = (unsigned*)(ws + 65536 + 256);      // 256 B
    unsigned long long* g_cand = (unsigned long long*)(ws + 66048);  // 64*4096*8 = 2 MB

    zero_kernel<<<(BQ * NBINS + 255) / 256, 256, 0, stream>>>(g_hist, BQ * NBINS);
    score_kernel<0><<<NB_BLK, 256, 0, stream>>>(query, vectors, g_hist, g_thr, g_cnt, g_cand);
    threshold_kernel<<<1, 64, 0, stream>>>(g_hist, g_thr, g_cnt);
    score_kernel<1><<<NB_BLK, 256, 0, stream>>>(query, vectors, g_hist, g_thr, g_cnt, g_cand);
    topk_kernel<<<BQ, 256, 0, stream>>>(g_cnt, g_cand, out_scores, out_idx);
}